// DocREModel_38774964748980
// MI455X (gfx1250) — compile-verified
//
#include <hip/hip_runtime.h>
#include <math.h>

// ---- problem constants (from reference) ----
#define B_     4
#define L_     1024
#define D_     768
#define H_     12
#define E_     32
#define M_     4
#define R_     512
#define NE_    256
#define HEADS_ 4
#define OC_    192
#define LAYERS_ 2
#define EMB_   768
#define BS_    64
#define NL_    97
#define NEDGE  (NE_ + E_)     // 288 edges incl. self loops
#define N_     (B_ * R_)      // 2048 pair rows
#define KBIL   (EMB_ * BS_)   // 49152 bilinear K

typedef __attribute__((ext_vector_type(2))) float v2f;
typedef __attribute__((ext_vector_type(8))) float v8f;

// CDNA5 fp32 WMMA: D(16x16) = A(16x4) * B(4x16) + C
__device__ __forceinline__ v8f wmma_f32_4(v2f a, v2f b, v8f c) {
  return __builtin_amdgcn_wmma_f32_16x16x4_f32(
      /*neg_a=*/false, a, /*neg_b=*/false, b,
      /*c_mod=*/(short)0, c, /*reuse_a=*/false, /*reuse_b=*/false);
}

// ---------------------------------------------------------------------------
// 1) entity embeddings: logsumexp over M gathered token rows
// ---------------------------------------------------------------------------
__global__ void k_entity_emb(const float* __restrict__ seq,
                             const int* __restrict__ starts,
                             float* __restrict__ e_emb) {
  int be = blockIdx.x;              // b*E + e
  int b = be / E_;
  int idx[M_];
#pragma unroll
  for (int m = 0; m < M_; ++m) idx[m] = starts[(size_t)be * M_ + m] + 1;
  const float* sb = seq + (size_t)b * L_ * D_;
  for (int d = threadIdx.x; d < D_; d += blockDim.x) {
    float v[M_], mx = -INFINITY;
#pragma unroll
    for (int m = 0; m < M_; ++m) {
      v[m] = sb[(size_t)idx[m] * D_ + d];
      mx = fmaxf(mx, v[m]);
    }
    float s = 0.f;
#pragma unroll
    for (int m = 0; m < M_; ++m) s += __expf(v[m] - mx);
    e_emb[(size_t)be * D_ + d] = mx + __logf(s);
  }
}

// ---------------------------------------------------------------------------
// 2) entity attention: e_att[b,e,h,l] = mean_m attention[b,h,starts+1,l]
// ---------------------------------------------------------------------------
__global__ void k_entity_att(const float* __restrict__ att,
                             const int* __restrict__ starts,
                             float* __restrict__ e_att) {
  int id = blockIdx.x;              // (b*E + e)*H + h
  int h = id % H_;
  int be = id / H_;
  int b = be / E_;
  int idx[M_];
#pragma unroll
  for (int m = 0; m < M_; ++m) idx[m] = starts[(size_t)be * M_ + m] + 1;
  const float* ab = att + ((size_t)b * H_ + h) * L_ * L_;
  const float inv = 1.0f / (float)M_;
  for (int l = threadIdx.x; l < L_; l += blockDim.x) {
    float s = 0.f;
#pragma unroll
    for (int m = 0; m < M_; ++m) s += ab[(size_t)idx[m] * L_ + l];
    e_att[((size_t)be * H_ + h) * L_ + l] = s * inv;
  }
}

// ---------------------------------------------------------------------------
// 3a) GAT: xw = x @ W   (per (b,e) row; W is 768x768, L2-resident)
// ---------------------------------------------------------------------------
__global__ void k_gat_xw(const float* __restrict__ xin,
                         const float* __restrict__ W,
                         float* __restrict__ xw) {
  __shared__ float sx[D_];
  int be = blockIdx.x;
  const float* xr = xin + (size_t)be * D_;
  for (int d = threadIdx.x; d < D_; d += blockDim.x) sx[d] = xr[d];
  __syncthreads();
  for (int o = threadIdx.x; o < D_; o += blockDim.x) {
    float acc = 0.f;
    for (int d = 0; d < D_; ++d) acc += sx[d] * W[(size_t)d * D_ + o];
    xw[(size_t)be * D_ + o] = acc;
  }
}

// ---------------------------------------------------------------------------
// 3b) GAT attention logits: al_s/al_d[b,e,h] = sum_oc xw * a_src/a_dst
// ---------------------------------------------------------------------------
__global__ void k_gat_al(const float* __restrict__ xw,
                         const float* __restrict__ a_src,
                         const float* __restrict__ a_dst,
                         float* __restrict__ al_s, float* __restrict__ al_d) {
  __shared__ float ps[128], pd[128];
  int be = blockIdx.x;
  int t = threadIdx.x;
  int h = t >> 5, lane = t & 31;
  const float* row = xw + (size_t)be * D_ + h * OC_;
  float ss = 0.f, sd = 0.f;
  for (int oc = lane; oc < OC_; oc += 32) {
    float v = row[oc];
    ss += v * a_src[h * OC_ + oc];
    sd += v * a_dst[h * OC_ + oc];
  }
  ps[t] = ss;
  pd[t] = sd;
  __syncthreads();
  if (lane == 0) {
    float as = 0.f, ad = 0.f;
    for (int i = 0; i < 32; ++i) { as += ps[h * 32 + i]; ad += pd[h * 32 + i]; }
    al_s[be * HEADS_ + h] = as;
    al_d[be * HEADS_ + h] = ad;
  }
}

// ---------------------------------------------------------------------------
// 3c) GAT edge softmax + aggregation + bias + ELU, one block per dst entity
// ---------------------------------------------------------------------------
__global__ void k_gat_edge(const float* __restrict__ xw,
                           const float* __restrict__ al_s,
                           const float* __restrict__ al_d,
                           const int* __restrict__ edge_index,
                           const float* __restrict__ bias,
                           float* __restrict__ xout) {
  __shared__ float ev[NEDGE][HEADS_];
  __shared__ int ssrc[NEDGE];
  __shared__ int smatch[NEDGE];
  int be = blockIdx.x;
  int b = be / E_, e = be % E_;
  const int* ei = edge_index + (size_t)b * 2 * NE_;
  for (int j = threadIdx.x; j < NEDGE; j += blockDim.x) {
    int sj = (j < NE_) ? ei[j] : (j - NE_);
    int dj = (j < NE_) ? ei[NE_ + j] : (j - NE_);
    ssrc[j] = sj;
    int match = (dj == e);
    smatch[j] = match;
#pragma unroll
    for (int h = 0; h < HEADS_; ++h) {
      float v = al_s[(b * E_ + sj) * HEADS_ + h] + al_d[(b * E_ + e) * HEADS_ + h];
      ev[j][h] = (v > 0.f) ? v : 0.2f * v;   // leaky_relu(0.2)
    }
  }
  __syncthreads();
  if (threadIdx.x < HEADS_) {               // deterministic serial softmax per head
    int h = threadIdx.x;
    float mx = -INFINITY;
    for (int j = 0; j < NEDGE; ++j)
      if (smatch[j]) mx = fmaxf(mx, ev[j][h]);
    float den = 0.f;
    for (int j = 0; j < NEDGE; ++j)
      if (smatch[j]) den += __expf(ev[j][h] - mx);
    float rden = 1.0f / (den + 1e-16f);
    for (int j = 0; j < NEDGE; ++j)
      ev[j][h] = smatch[j] ? __expf(ev[j][h] - mx) * rden : 0.f;
  }
  __syncthreads();
  for (int o = threadIdx.x; o < D_; o += blockDim.x) {
    int h = o / OC_;
    float acc = 0.f;
    for (int j = 0; j < NEDGE; ++j) {
      float a = ev[j][h];
      if (a != 0.f) acc += a * xw[(size_t)(b * E_ + ssrc[j]) * D_ + o];
    }
    acc += bias[o];
    xout[(size_t)be * D_ + o] = (acc > 0.f) ? acc : (__expf(acc) - 1.0f);  // ELU
  }
}

// ---------------------------------------------------------------------------
// 4) ht_att[b,r,l] = normalize_l( mean_h e_att[h_idx]*e_att[t_idx] )
// ---------------------------------------------------------------------------
__global__ void k_ht_att(const float* __restrict__ e_att,
                         const int* __restrict__ hts,
                         float* __restrict__ ht_att) {
  __shared__ float red[256];
  int br = blockIdx.x;                       // b*R + r
  int b = br / R_;
  int hi = hts[br * 2 + 0], ti = hts[br * 2 + 1];
  const float* ha = e_att + (size_t)(b * E_ + hi) * H_ * L_;
  const float* ta = e_att + (size_t)(b * E_ + ti) * H_ * L_;
  float v[4], psum = 0.f;
#pragma unroll
  for (int q = 0; q < 4; ++q) {
    int l = threadIdx.x + q * 256;
    float s = 0.f;
#pragma unroll
    for (int h = 0; h < H_; ++h) s += ha[h * L_ + l] * ta[h * L_ + l];
    v[q] = s * (1.0f / (float)H_);
    psum += v[q];
  }
  red[threadIdx.x] = psum;
  __syncthreads();
  for (int off = 128; off > 0; off >>= 1) {
    if (threadIdx.x < off) red[threadIdx.x] += red[threadIdx.x + off];
    __syncthreads();
  }
  float rnorm = 1.0f / (red[0] + 1e-30f);
#pragma unroll
  for (int q = 0; q < 4; ++q) {
    int l = threadIdx.x + q * 256;
    ht_att[(size_t)br * L_ + l] = v[q] * rnorm;
  }
}

// ---------------------------------------------------------------------------
// 5) rs[b] = ht_att[b] (512x1024) @ seq[b] (1024x768)  -- fp32 WMMA
//    one wave per 16x64 strip (4 N-tiles), A fragment reused 4x per K-step
//    grid (D/64, R/16, B) = 1536 waves
// ---------------------------------------------------------------------------
__global__ void k_rs(const float* __restrict__ htA,
                     const float* __restrict__ seq,
                     float* __restrict__ rs) {
  int b = blockIdx.z;
  int m0 = blockIdx.y * 16;
  int n0 = blockIdx.x * 64;
  const float* A = htA + (size_t)b * R_ * L_;
  const float* Bm = seq + (size_t)b * L_ * D_;
  int lane = threadIdx.x;
  int mi = lane & 15;
  int kd = (lane < 16) ? 0 : 2;
  v8f acc0 = {}, acc1 = {}, acc2 = {}, acc3 = {};
#pragma unroll 2
  for (int k0 = 0; k0 < L_; k0 += 4) {
    const float* ar = A + (size_t)(m0 + mi) * L_ + k0 + kd;
    v2f a;
    a.x = ar[0];
    a.y = ar[1];
    const float* w0 = Bm + (size_t)(k0 + kd) * D_ + n0 + mi;
    const float* w1 = w0 + D_;
    v2f b0, b1, b2, b3;
    b0.x = w0[0];  b0.y = w1[0];
    b1.x = w0[16]; b1.y = w1[16];
    b2.x = w0[32]; b2.y = w1[32];
    b3.x = w0[48]; b3.y = w1[48];
    acc0 = wmma_f32_4(a, b0, acc0);
    acc1 = wmma_f32_4(a, b1, acc1);
    acc2 = wmma_f32_4(a, b2, acc2);
    acc3 = wmma_f32_4(a, b3, acc3);
  }
  int rbase = (lane < 16) ? 0 : 8;
#pragma unroll
  for (int r = 0; r < 8; ++r) {
    float* orow = rs + ((size_t)b * R_ + m0 + rbase + r) * D_ + n0 + mi;
    orow[0]  = acc0[r];
    orow[16] = acc1[r];
    orow[32] = acc2[r];
    orow[48] = acc3[r];
  }
}

// ---------------------------------------------------------------------------
// 6) build hcat = [x_gat[h_idx] | rs], tcat = [x_gat[t_idx] | rs]
// ---------------------------------------------------------------------------
__global__ void k_concat(const float* __restrict__ xg,
                         const float* __restrict__ rsB,
                         const int* __restrict__ hts,
                         float* __restrict__ hcat, float* __restrict__ tcat) {
  size_t total = (size_t)N_ * (2 * D_);
  for (size_t i = (size_t)blockIdx.x * blockDim.x + threadIdx.x; i < total;
       i += (size_t)gridDim.x * blockDim.x) {
    int n = (int)(i / (2 * D_));
    int c = (int)(i % (2 * D_));
    int b = n / R_;
    if (c < D_) {
      int hi = hts[n * 2 + 0], ti = hts[n * 2 + 1];
      hcat[i] = xg[(size_t)(b * E_ + hi) * D_ + c];
      tcat[i] = xg[(size_t)(b * E_ + ti) * D_ + c];
    } else {
      float v = rsB[(size_t)n * D_ + (c - D_)];
      hcat[i] = v;
      tcat[i] = v;
    }
  }
}

// ---------------------------------------------------------------------------
// 7) C = tanh(A (2048x1536) @ W (1536x768) + bias)  -- fp32 WMMA
//    one wave per 16x64 strip (4 N-tiles); grid (D/64, N_/16) = 1536 waves
// ---------------------------------------------------------------------------
__global__ void k_gemm_tanh(const float* __restrict__ A,
                            const float* __restrict__ W,
                            const float* __restrict__ bias,
                            float* __restrict__ C) {
  int m0 = blockIdx.y * 16;
  int n0 = blockIdx.x * 64;
  int lane = threadIdx.x;
  int mi = lane & 15;
  int kd = (lane < 16) ? 0 : 2;
  const int K = 2 * D_;
  v8f acc0 = {}, acc1 = {}, acc2 = {}, acc3 = {};
#pragma unroll 2
  for (int k0 = 0; k0 < K; k0 += 4) {
    const float* ar = A + (size_t)(m0 + mi) * K + k0 + kd;
    v2f a;
    a.x = ar[0];
    a.y = ar[1];
    const float* w0 = W + (size_t)(k0 + kd) * D_ + n0 + mi;
    const float* w1 = w0 + D_;
    v2f b0, b1, b2, b3;
    b0.x = w0[0];  b0.y = w1[0];
    b1.x = w0[16]; b1.y = w1[16];
    b2.x = w0[32]; b2.y = w1[32];
    b3.x = w0[48]; b3.y = w1[48];
    acc0 = wmma_f32_4(a, b0, acc0);
    acc1 = wmma_f32_4(a, b1, acc1);
    acc2 = wmma_f32_4(a, b2, acc2);
    acc3 = wmma_f32_4(a, b3, acc3);
  }
  int rbase = (lane < 16) ? 0 : 8;
  float bv0 = bias[n0 + mi];
  float bv1 = bias[n0 + 16 + mi];
  float bv2 = bias[n0 + 32 + mi];
  float bv3 = bias[n0 + 48 + mi];
#pragma unroll
  for (int r = 0; r < 8; ++r) {
    float* orow = C + (size_t)(m0 + rbase + r) * D_ + n0 + mi;
    orow[0]  = tanhf(acc0[r] + bv0);
    orow[16] = tanhf(acc1[r] + bv1);
    orow[32] = tanhf(acc2[r] + bv2);
    orow[48] = tanhf(acc3[r] + bv3);
  }
}

// ---------------------------------------------------------------------------
// 8) logits = (b1 outer b2) @ bil_W + bil_b, A generated on the fly:
//    A[n,kappa] = hs[n, kappa>>6] * ts[n, (kappa>>12)*64 + (kappa&63)]
//    block: 16 rows, 8 waves (waves 0..6 = one 16-col N tile each, 97->112)
//    ts tile staged in LDS (48KB); hs scalar reused across 16 WMMA steps.
// ---------------------------------------------------------------------------
__global__ void k_bilinear(const float* __restrict__ hs,
                           const float* __restrict__ ts,
                           const float* __restrict__ bW,
                           const float* __restrict__ bilb,
                           float* __restrict__ out) {
  __shared__ float sts[16][D_ + 1];
  int m0 = blockIdx.x * 16;
  int t = threadIdx.x;
  for (int i = t; i < 16 * D_; i += blockDim.x) {
    int m = i / D_, c = i % D_;
    sts[m][c] = ts[(size_t)(m0 + m) * D_ + c];
  }
  __syncthreads();
  int wave = t >> 5;
  int lane = t & 31;
  if (wave >= 7) return;                    // wave-uniform; active waves keep EXEC all-ones
  int n0 = wave * 16;
  int mi = lane & 15;
  int kd = (lane < 16) ? 0 : 2;
  int col = n0 + mi;
  int colc = (col < NL_) ? col : (NL_ - 1); // clamp loads, mask store
  const float* hrow = hs + (size_t)(m0 + mi) * D_;
  v8f acc = {};
  // kc = kappa>>6 = k*64+i ; inner j covers 64 kappa = 16 WMMA steps
  for (int kc = 0; kc < EMB_; ++kc) {
    float hv = hrow[kc];                    // one global scalar per 16 WMMAs
    int tbase = kc & ~63;                   // k*64
    const float* wB = bW + ((size_t)kc << 6) * NL_ + colc;
#pragma unroll 4
    for (int jj = 0; jj < 64; jj += 4) {
      v2f a, bb;
      a.x = hv * sts[mi][tbase + jj + kd];
      a.y = hv * sts[mi][tbase + jj + kd + 1];
      bb.x = wB[(size_t)(jj + kd) * NL_];
      bb.y = wB[(size_t)(jj + kd + 1) * NL_];
      acc = wmma_f32_4(a, bb, acc);
    }
  }
  if (col < NL_) {
    int rbase = (lane < 16) ? 0 : 8;
    float bv = bilb[col];
#pragma unroll
    for (int r = 0; r < 8; ++r)
      out[(size_t)(m0 + rbase + r) * NL_ + col] = acc[r] + bv;
  }
}

// ---------------------------------------------------------------------------
extern "C" void kernel_launch(void* const* d_in, const int* in_sizes, int n_in,
                              void* d_out, int out_size, void* d_ws, size_t ws_size,
                              hipStream_t stream) {
  const float* seq    = (const float*)d_in[0];
  const float* att    = (const float*)d_in[1];
  const int*   starts = (const int*)d_in[2];
  const int*   hts    = (const int*)d_in[3];
  const int*   eidx   = (const int*)d_in[4];
  const float* gat_W  = (const float*)d_in[5];
  const float* a_src  = (const float*)d_in[6];
  const float* a_dst  = (const float*)d_in[7];
  const float* gat_b  = (const float*)d_in[8];
  const float* head_W = (const float*)d_in[9];
  const float* head_b = (const float*)d_in[10];
  const float* tail_W = (const float*)d_in[11];
  const float* tail_b = (const float*)d_in[12];
  const float* bil_W  = (const float*)d_in[13];
  const float* bil_b  = (const float*)d_in[14];
  float* out = (float*)d_out;

  // carve workspace (~61 MB of fp32)
  float* p = (float*)d_ws;
  float* e_emb = p; p += (size_t)B_ * E_ * D_;
  float* e_att = p; p += (size_t)B_ * E_ * H_ * L_;
  float* xw    = p; p += (size_t)B_ * E_ * D_;
  float* als   = p; p += (size_t)B_ * E_ * HEADS_;
  float* ald   = p; p += (size_t)B_ * E_ * HEADS_;
  float* xa    = p; p += (size_t)B_ * E_ * D_;
  float* xb    = p; p += (size_t)B_ * E_ * D_;
  float* htw   = p; p += (size_t)B_ * R_ * L_;
  float* rsw   = p; p += (size_t)B_ * R_ * D_;
  float* hcat  = p; p += (size_t)N_ * 2 * D_;
  float* tcat  = p; p += (size_t)N_ * 2 * D_;
  float* hs2   = p; p += (size_t)N_ * D_;
  float* ts2   = p; p += (size_t)N_ * D_;

  k_entity_emb<<<B_ * E_, 256, 0, stream>>>(seq, starts, e_emb);
  k_entity_att<<<B_ * E_ * H_, 256, 0, stream>>>(att, starts, e_att);

  const float* xin = e_emb;
  float* bufs[2] = {xa, xb};
  for (int l = 0; l < LAYERS_; ++l) {
    k_gat_xw<<<B_ * E_, 256, 0, stream>>>(xin, gat_W + (size_t)l * D_ * D_, xw);
    k_gat_al<<<B_ * E_, 128, 0, stream>>>(xw, a_src + l * HEADS_ * OC_,
                                          a_dst + l * HEADS_ * OC_, als, ald);
    k_gat_edge<<<B_ * E_, 256, 0, stream>>>(xw, als, ald, eidx,
                                            gat_b + l * D_, bufs[l]);
    xin = bufs[l];
  }

  k_ht_att<<<B_ * R_, 256, 0, stream>>>(e_att, hts, htw);
  k_rs<<<dim3(D_ / 64, R_ / 16, B_), 32, 0, stream>>>(htw, seq, rsw);
  k_concat<<<1024, 256, 0, stream>>>(xin, rsw, hts, hcat, tcat);
  k_gemm_tanh<<<dim3(D_ / 64, N_ / 16), 32, 0, stream>>>(hcat, head_W, head_b, hs2);
  k_gemm_tanh<<<dim3(D_ / 64, N_ / 16), 32, 0, stream>>>(tcat, tail_W, tail_b, ts2);
  k_bilinear<<<N_ / 16, 256, 0, stream>>>(hs2, ts2, bil_W, bil_b, out);
}